// GCN_21165598834931
// MI455X (gfx1250) — compile-verified
//
#include <hip/hip_runtime.h>

// ---------------------------------------------------------------------------
// GCN (2-layer, GCNConv with self-loops + symmetric norm) for MI455X / gfx1250
//
// Dense parts use V_WMMA_F32_16X16X4_F32 (full fp32 precision; GEMM flops are
// negligible vs. the L2-resident edge aggregation, so no need for bf16/fp8).
// Edge aggregation: wave32 per edge, coalesced float4/float2 gather of the
// transformed source row, global_atomic_add_f32 scatter (tables fit in 192MB L2).
// Self-loop + bias folded into the accumulator initializer; ReLU fused into
// GEMM2's A-operand load.
// ---------------------------------------------------------------------------

typedef __attribute__((ext_vector_type(2))) float v2f;
typedef __attribute__((ext_vector_type(8))) float v8f;

// ---------------- small elementwise kernels ----------------

__global__ void k_fill_f32(float* __restrict__ p, float v, int n) {
    int i = blockIdx.x * blockDim.x + threadIdx.x;
    if (i < n) p[i] = v;
}

__global__ void k_deg_count(const int* __restrict__ dst, float* __restrict__ deg, int E) {
    int i = blockIdx.x * blockDim.x + threadIdx.x;
    if (i < E) atomicAdd(&deg[dst[i]], 1.0f);
}

__global__ void k_rsqrt_inplace(float* __restrict__ p, int n) {
    int i = blockIdx.x * blockDim.x + threadIdx.x;
    if (i < n) p[i] = rsqrtf(p[i]);   // deg >= 1 always (self-loop), no zero case
}

// out[i*C + c] = bias[c] + dinv[i]^2 * h[i*C + c]   (C = 1<<shift)
__global__ void k_self_loop_bias(float* __restrict__ out, const float* __restrict__ h,
                                 const float* __restrict__ bias, const float* __restrict__ dinv,
                                 int total, int shift) {
    int idx = blockIdx.x * blockDim.x + threadIdx.x;
    if (idx >= total) return;
    int r = idx >> shift;
    int c = idx & ((1 << shift) - 1);
    float di = dinv[r];
    out[idx] = bias[c] + di * di * h[idx];
}

// ---------------- edge scatter kernels (wave32 per edge) ----------------

// 128 channels: lane l handles channels 4l..4l+3 (one float4 gather, 4 atomics)
__global__ void k_edge_scatter_128(const int* __restrict__ src, const int* __restrict__ dst,
                                   const float* __restrict__ dinv, const float* __restrict__ h,
                                   float* __restrict__ agg, int E) {
    int gid  = blockIdx.x * blockDim.x + threadIdx.x;
    int e    = gid >> 5;
    int lane = gid & 31;
    if (e >= E) return;
    int s = src[e];
    int d = dst[e];
    float w = dinv[s] * dinv[d];
    float4 v = ((const float4*)(h + (size_t)s * 128))[lane];
    float* o = agg + (size_t)d * 128 + lane * 4;
    atomicAdd(o + 0, w * v.x);
    atomicAdd(o + 1, w * v.y);
    atomicAdd(o + 2, w * v.z);
    atomicAdd(o + 3, w * v.w);
}

// 64 channels: lane l handles channels 2l..2l+1 (one float2 gather, 2 atomics)
__global__ void k_edge_scatter_64(const int* __restrict__ src, const int* __restrict__ dst,
                                  const float* __restrict__ dinv, const float* __restrict__ h,
                                  float* __restrict__ agg, int E) {
    int gid  = blockIdx.x * blockDim.x + threadIdx.x;
    int e    = gid >> 5;
    int lane = gid & 31;
    if (e >= E) return;
    int s = src[e];
    int d = dst[e];
    float w = dinv[s] * dinv[d];
    float2 v = ((const float2*)(h + (size_t)s * 64))[lane];
    float* o = agg + (size_t)d * 64 + lane * 2;
    atomicAdd(o + 0, w * v.x);
    atomicAdd(o + 1, w * v.y);
}

// ---------------- WMMA fp32 GEMM:  C[nrows,OUT] = act(A)[nrows,128] @ W[OUT,128]^T
//
// Per-wave: one 16-row strip, OUT/16 column tiles, K=128 in steps of 4 via
// V_WMMA_F32_16X16X4_F32. Operand layouts per CDNA5 ISA 7.12.2 (32-bit):
//   A 16x4 : lane&15 = M,  (lane>>4)*2 = K-pair base, 2 VGPRs (K, K+1)
//   B 4x16 : lane&15 = N,  (lane>>4)*2 = K-pair base, 2 VGPRs (K, K+1)
//   C 16x16: lane&15 = N,  VGPR r -> M = r + 8*(lane>>4)
// Wave-uniform guard only (EXEC all-ones across every WMMA, as required).
// ----------------

template <int OUT, bool RELU>
__global__ void k_gemm_wmma_f32(const float* __restrict__ A, const float* __restrict__ W,
                                float* __restrict__ C, int nrows) {
    constexpr int K  = 128;
    constexpr int NT = OUT / 16;

    const int lane  = threadIdx.x & 31;
    const int wave  = threadIdx.x >> 5;
    const int mtile = blockIdx.x * (blockDim.x >> 5) + wave;
    const int row0  = mtile * 16;
    if (row0 >= nrows) return;                     // wave-uniform exit

    v8f acc[NT] = {};                              // C starts at 0 (bias folded later)

    const int half2  = (lane >> 4) * 2;            // 0 or 2: K offset for this lane half
    const int mlocal = lane & 15;
    const float2* __restrict__ Arow = (const float2*)(A + (size_t)(row0 + mlocal) * K);
    const float2* __restrict__ Wp   = (const float2*)W;

    for (int k0 = 0; k0 < K; k0 += 4) {
        const int kb = k0 + half2;                 // even -> float2 aligned
        float2 av = Arow[kb >> 1];
        if (RELU) {
            av.x = fmaxf(av.x, 0.0f);
            av.y = fmaxf(av.y, 0.0f);
        }
        v2f a;
        a.x = av.x;
        a.y = av.y;
#pragma unroll
        for (int t = 0; t < NT; ++t) {
            const int col = t * 16 + mlocal;       // B[k][col] = W[col*K + k]
            float2 bv = Wp[((size_t)col * K + kb) >> 1];
            v2f b;
            b.x = bv.x;
            b.y = bv.y;
            acc[t] = __builtin_amdgcn_wmma_f32_16x16x4_f32(
                /*neg_a=*/false, a, /*neg_b=*/false, b,
                /*c_mod=*/(short)0, acc[t], /*reuse_a=*/false, /*reuse_b=*/false);
        }
    }

    // store: VGPR r holds row (row0 + 8*(lane>>4) + r), col = t*16 + (lane&15)
    const int crow = row0 + (lane >> 4) * 8;
#pragma unroll
    for (int t = 0; t < NT; ++t) {
#pragma unroll
        for (int r = 0; r < 8; ++r) {
            C[(size_t)(crow + r) * OUT + t * 16 + mlocal] = acc[t][r];
        }
    }
}

// ---------------- launcher ----------------

extern "C" void kernel_launch(void* const* d_in, const int* in_sizes, int n_in,
                              void* d_out, int out_size, void* d_ws, size_t ws_size,
                              hipStream_t stream) {
    const float* x  = (const float*)d_in[0];   // [N,128]
    const int*   ei = (const int*)  d_in[1];   // [2,E]
    const float* W1 = (const float*)d_in[2];   // [128,128]
    const float* b1 = (const float*)d_in[3];   // [128]
    const float* W2 = (const float*)d_in[4];   // [64,128]
    const float* b2 = (const float*)d_in[5];   // [64]
    float* out = (float*)d_out;                // [N,64]

    const int N = in_sizes[0] / 128;
    const int E = in_sizes[1] / 2;
    const int* src = ei;        // edge_index[0]
    const int* dst = ei + E;    // edge_index[1]

    // workspace layout (floats): dinv[N] | h1[N*128] | agg1[N*128] | h2[N*64]
    float* dinv = (float*)d_ws;
    float* h1   = dinv + (size_t)N;
    float* agg1 = h1 + (size_t)N * 128;
    float* h2   = agg1 + (size_t)N * 128;

    const int T = 256;

    // 1) degree (with self-loop) -> dinv = rsqrt(deg)
    k_fill_f32<<<(N + T - 1) / T, T, 0, stream>>>(dinv, 1.0f, N);
    k_deg_count<<<(E + T - 1) / T, T, 0, stream>>>(dst, dinv, E);
    k_rsqrt_inplace<<<(N + T - 1) / T, T, 0, stream>>>(dinv, N);

    // 2) h1 = x @ W1^T   (fp32 WMMA)
    {
        int mtiles = (N + 15) / 16;
        int blocks = (mtiles + 7) / 8;             // 8 waves/block, 16 rows/wave
        k_gemm_wmma_f32<128, false><<<blocks, T, 0, stream>>>(x, W1, h1, N);
    }

    // 3) agg1 = b1 + dinv^2 * h1  (self-loop + bias), then scatter E edges
    k_self_loop_bias<<<((size_t)N * 128 + T - 1) / T, T, 0, stream>>>(agg1, h1, b1, dinv, N * 128, 7);
    k_edge_scatter_128<<<((size_t)E * 32 + T - 1) / T, T, 0, stream>>>(src, dst, dinv, h1, agg1, E);

    // 4) h2 = relu(agg1) @ W2^T  (ReLU fused into A load)
    {
        int mtiles = (N + 15) / 16;
        int blocks = (mtiles + 7) / 8;
        k_gemm_wmma_f32<64, true><<<blocks, T, 0, stream>>>(agg1, W2, h2, N);
    }

    // 5) out = b2 + dinv^2 * h2, then scatter E edges
    k_self_loop_bias<<<((size_t)N * 64 + T - 1) / T, T, 0, stream>>>(out, h2, b2, dinv, N * 64, 6);
    k_edge_scatter_64<<<((size_t)E * 32 + T - 1) / T, T, 0, stream>>>(src, dst, dinv, h2, out, E);
}